// ES_40527311405627
// MI455X (gfx1250) — compile-verified
//
#include <hip/hip_runtime.h>
#include <stdint.h>

typedef __attribute__((ext_vector_type(4))) float v4f;

#define N_SERIES 512
#define N_TIME   1024
#define INSZ     168
#define OUTSZ    24
#define SM       24          // seasonality
#define N_WIN    833         // N_TIME - INSZ - OUTSZ + 1
#define SEAS_LEN 1048        // SM + 1 + (N_TIME-1)
#define WT       64          // windows per block in expansion kernel

// d_out layout (flat, return order): y_hat, y, levels, seas
#define OFF_Y    ((size_t)N_WIN * N_SERIES * INSZ)                 // 71,651,328
#define OFF_LEV  (OFF_Y + (size_t)N_WIN * N_SERIES * OUTSZ)       // 81,887,232
#define OFF_SEAS (OFF_LEV + (size_t)N_SERIES * N_TIME)            // 82,411,520

// ---------------------------------------------------------------------------
// Kernel 1: per-series ES recurrence. Delay-24 seasonal buffer kept in 24
// VGPRs. 1023 steps = 42 * 24 + 15: outer loop of 42 constant-trip,
// break-free, fully-unrolled 24-phase blocks (static ring indices -> VGPRs),
// then a statically unrolled 15-step tail (phase restarts at 0: 1008%24==0).
// ---------------------------------------------------------------------------
__global__ __launch_bounds__(128)
void k_recur(const float* __restrict__ y, const int* __restrict__ idxs,
             const float* __restrict__ ew,
             float* __restrict__ lev, float* __restrict__ seas)
{
    int s = blockIdx.x * blockDim.x + threadIdx.x;
    if (s >= N_SERIES) return;
    const float* e = ew + (size_t)idxs[s] * (2 + SM);
    float a = 1.0f / (1.0f + expf(-e[0]));   // lev_sms
    float b = 1.0f / (1.0f + expf(-e[1]));   // seas_sms

    float ring[SM];
    float* srow = seas + (size_t)s * SEAS_LEN;
    float is0 = expf(e[2]);
    srow[0]  = is0;          // seasonalities[0]   = init_seas[0]
    srow[SM] = is0;          // seasonalities[24]  = init_seas[0]
    ring[SM - 1] = is0;      // buf0 = [init[1..23], init[0]]
#pragma unroll
    for (int j = 1; j < SM; ++j) {
        float v = expf(e[2 + j]);
        srow[j] = v;
        ring[j - 1] = v;
    }

    const float* yr = y + (size_t)s * N_TIME;
    float* lr = lev + (size_t)s * N_TIME;
    float levv = yr[0] / is0;
    lr[0] = levv;

    int t = 1;
    auto step = [&](int k) {
        float yt = yr[t];
        float st = ring[k];              // seasonalities[t] (delay-24)
        float nl = a * (yt / st) + (1.0f - a) * levv;
        float ns = b * (yt / nl) + (1.0f - b) * st;
        ring[k] = ns;
        lr[t] = nl;
        srow[SM + t] = ns;               // seasonalities[24 + t]
        levv = nl;
        ++t;
    };

    for (int blk = 0; blk < 42; ++blk) { // 42 full seasonal cycles
#pragma unroll
        for (int k = 0; k < SM; ++k) step(k);
    }
#pragma unroll
    for (int k = 0; k < 15; ++k) step(k);  // tail: t = 1009..1023
}

// ---------------------------------------------------------------------------
// Kernel 2: massively parallel log precompute into workspace.
//   L[s,t]      = log(y[s,t]) - log(seas[s,t])
//   loglev[s,t] = log(levels[s,t])
// ---------------------------------------------------------------------------
__global__ __launch_bounds__(256)
void k_logs(const float* __restrict__ y, const float* __restrict__ lev,
            const float* __restrict__ seas,
            float* __restrict__ L, float* __restrict__ LL)
{
    int id = blockIdx.x * blockDim.x + threadIdx.x;
    if (id >= N_SERIES * N_TIME) return;
    int s = id >> 10;
    int t = id & (N_TIME - 1);
    L[id]  = logf(y[id]) - logf(seas[(size_t)s * SEAS_LEN + t]);
    LL[id] = logf(lev[id]);
}

// ---------------------------------------------------------------------------
// Kernel 3: store-bound window expansion. One block = one series x 64
// windows. Stage L[w0..w0+255] and loglev[w0+160..w0+287] into LDS with
// CDNA5 async-to-LDS b128 ops (ASYNCcnt path), then NT-store float4 rows.
//   y_hat[w,s,i] = L[s,w+i]     - loglev[s,w+167]
//   y    [w,s,o] = L[s,w+168+o] - loglev[s,w+168]
// ---------------------------------------------------------------------------
__global__ __launch_bounds__(256)
void k_windows(const float* __restrict__ L, const float* __restrict__ LL,
               float* __restrict__ yhat, float* __restrict__ yout)
{
    __shared__ float sL[256];   // L[s, w0 .. w0+255]
    __shared__ float sV[128];   // loglev[s, w0+160 .. w0+287]

    int tid = threadIdx.x;
    int w0  = blockIdx.x * WT;
    int s   = blockIdx.y;
    const float* Lrow = L  + (size_t)s * N_TIME;
    const float* Vrow = LL + (size_t)s * N_TIME;

    // 96 lanes (waves 0..2) each async-copy one 16B chunk: 64 for sL, 32 for sV.
    if (tid < 96) {
        uint64_t ga;
        unsigned lofs;
        if (tid < 64) {
            int idx = w0 + 4 * tid;
            if (idx > N_TIME - 4) idx = N_TIME - 4;   // clamp (clamped chunks unused)
            ga   = (uint64_t)(uintptr_t)(Lrow + idx);
            lofs = (unsigned)(uintptr_t)(&sL[4 * tid]);   // low 32b of LDS ptr = LDS offset
        } else {
            int u = tid - 64;
            int idx = w0 + 160 + 4 * u;
            if (idx > N_TIME - 4) idx = N_TIME - 4;
            ga   = (uint64_t)(uintptr_t)(Vrow + idx);
            lofs = (unsigned)(uintptr_t)(&sV[4 * u]);
        }
        asm volatile("global_load_async_to_lds_b128 %0, %1, off"
                     :: "v"(lofs), "v"(ga) : "memory");
    }
#if __has_builtin(__builtin_amdgcn_s_wait_asynccnt)
    __builtin_amdgcn_s_wait_asynccnt(0);
#else
    asm volatile("s_wait_asynccnt 0" ::: "memory");
#endif
    __syncthreads();

    int nW = N_WIN - w0;
    if (nW > WT) nW = WT;

    // windows_y_hat: nW * 42 float4 stores
    int nq = nW * (INSZ / 4);
    for (int j = tid; j < nq; j += 256) {
        int wl = j / 42;
        int i  = (j - wl * 42) * 4;
        float base = sV[wl + 7];              // loglev[s, w+167]
        v4f v;
        v.x = sL[wl + i]     - base;
        v.y = sL[wl + i + 1] - base;
        v.z = sL[wl + i + 2] - base;
        v.w = sL[wl + i + 3] - base;
        size_t o = ((size_t)(w0 + wl) * N_SERIES + s) * INSZ + i;
        __builtin_nontemporal_store(v, (v4f*)(yhat + o));
    }

    // windows_y: nW * 6 float4 stores
    int nq2 = nW * (OUTSZ / 4);
    for (int j = tid; j < nq2; j += 256) {
        int wl = j / 6;
        int i  = (j - wl * 6) * 4;
        float base = sV[wl + 8];              // loglev[s, w+168]
        v4f v;
        v.x = sL[wl + INSZ + i]     - base;
        v.y = sL[wl + INSZ + i + 1] - base;
        v.z = sL[wl + INSZ + i + 2] - base;
        v.w = sL[wl + INSZ + i + 3] - base;
        size_t o = ((size_t)(w0 + wl) * N_SERIES + s) * OUTSZ + i;
        __builtin_nontemporal_store(v, (v4f*)(yout + o));
    }
}

// ---------------------------------------------------------------------------
extern "C" void kernel_launch(void* const* d_in, const int* in_sizes, int n_in,
                              void* d_out, int out_size, void* d_ws, size_t ws_size,
                              hipStream_t stream)
{
    (void)in_sizes; (void)n_in; (void)out_size; (void)ws_size;
    const float* y    = (const float*)d_in[0];
    const int*   idxs = (const int*)d_in[1];
    const float* ew   = (const float*)d_in[2];

    float* out  = (float*)d_out;
    float* yhat = out;
    float* yw   = out + OFF_Y;
    float* lev  = out + OFF_LEV;
    float* seas = out + OFF_SEAS;

    float* L  = (float*)d_ws;                       // 512*1024 f32
    float* LL = L + (size_t)N_SERIES * N_TIME;      // 512*1024 f32 (4 MB total)

    k_recur<<<dim3((N_SERIES + 127) / 128), dim3(128), 0, stream>>>(y, idxs, ew, lev, seas);
    k_logs<<<dim3((N_SERIES * N_TIME) / 256), dim3(256), 0, stream>>>(y, lev, seas, L, LL);
    k_windows<<<dim3((N_WIN + WT - 1) / WT, N_SERIES), dim3(256), 0, stream>>>(L, LL, yhat, yw);
}